// PlaceDistanceLoss_47132971107102
// MI455X (gfx1250) — compile-verified
//
#include <hip/hip_runtime.h>
#include <hip/hip_bf16.h>
#include <math.h>

// ---------------------------------------------------------------------------
// PlaceDistanceLoss for MI455X (gfx1250, wave32)
// N = 384 places, D = 512 embedding dim.
// sim GEMM uses V_WMMA_F32_16X16X4_F32 (f32 WMMA) to match f32 reference
// precision; everything else is small VALU work.
// ---------------------------------------------------------------------------

#define PDL_N   384
#define PDL_D   512
#define PDL_NN  (PDL_N * PDL_N)          // 147456
#define PDL_DBLK 576                      // dist kernel blocks: 576*256 == NN

typedef __attribute__((ext_vector_type(2))) float v2f;
typedef __attribute__((ext_vector_type(8))) float v8f;

// ---------------------------------------------------------------------------
// 1) Row norms of embeddings: norms[i] = ||E[i,:]||
// ---------------------------------------------------------------------------
__global__ void pdl_norms_kernel(const float* __restrict__ E,
                                 float* __restrict__ norms) {
    __shared__ float red[256];
    const int i = blockIdx.x;
    const int t = threadIdx.x;
    float s = 0.f;
    for (int k = t; k < PDL_D; k += 256) {
        float v = E[i * PDL_D + k];
        s += v * v;
    }
    red[t] = s;
    __syncthreads();
    for (int off = 128; off > 0; off >>= 1) {
        if (t < off) red[t] += red[t + off];
        __syncthreads();
    }
    if (t == 0) norms[i] = sqrtf(red[0]);
}

// ---------------------------------------------------------------------------
// 2) Haversine distance matrix + per-block partial sums (for sigma)
//    dist[i,j] = (2/pi) * atan2(sqrt(a), sqrt(1-a)),  a clipped to [1e-12, 1]
// ---------------------------------------------------------------------------
__global__ void pdl_dist_kernel(const float* __restrict__ coords,
                                float* __restrict__ dist,
                                float* __restrict__ partials) {
    const float DEG2RAD = 0.017453292519943295f;
    const float INV_PI  = 0.3183098861837907f;

    const int idx = blockIdx.x * 256 + threadIdx.x;   // exact cover: 576*256 == NN
    const int i = idx / PDL_N;
    const int j = idx - i * PDL_N;

    const float lat1 = coords[2 * i]     * DEG2RAD;
    const float lon1 = coords[2 * i + 1] * DEG2RAD;
    const float lat2 = coords[2 * j]     * DEG2RAD;
    const float lon2 = coords[2 * j + 1] * DEG2RAD;

    const float dlat = lat1 - lat2;
    const float dlon = lon1 - lon2;
    const float s1 = sinf(0.5f * dlat);
    const float s2 = sinf(0.5f * dlon);
    float a = s1 * s1 + cosf(lat1) * cosf(lat2) * s2 * s2;
    a = fminf(fmaxf(a, 1e-12f), 1.0f);
    const float c = 2.0f * atan2f(sqrtf(a), sqrtf(1.0f - a));
    const float d = c * INV_PI;

    dist[idx] = d;

    __shared__ float rs[256];
    __shared__ float rq[256];
    rs[threadIdx.x] = d;
    rq[threadIdx.x] = d * d;
    __syncthreads();
    for (int off = 128; off > 0; off >>= 1) {
        if (threadIdx.x < off) {
            rs[threadIdx.x] += rs[threadIdx.x + off];
            rq[threadIdx.x] += rq[threadIdx.x + off];
        }
        __syncthreads();
    }
    if (threadIdx.x == 0) {
        partials[2 * blockIdx.x]     = rs[0];
        partials[2 * blockIdx.x + 1] = rq[0];
    }
}

// ---------------------------------------------------------------------------
// 3) sigma = std(dist, ddof=1) over all N*N entries; store 1/(2*sigma^2)
// ---------------------------------------------------------------------------
__global__ void pdl_sigma_kernel(const float* __restrict__ partials,
                                 float* __restrict__ stats) {
    __shared__ double rs[256];
    __shared__ double rq[256];
    const int t = threadIdx.x;
    double s = 0.0, q = 0.0;
    for (int b = t; b < PDL_DBLK; b += 256) {
        s += (double)partials[2 * b];
        q += (double)partials[2 * b + 1];
    }
    rs[t] = s;
    rq[t] = q;
    __syncthreads();
    for (int off = 128; off > 0; off >>= 1) {
        if (t < off) { rs[t] += rs[t + off]; rq[t] += rq[t + off]; }
        __syncthreads();
    }
    if (t == 0) {
        const double nn = (double)PDL_NN;
        const double mean = rs[0] / nn;
        const double var  = (rq[0] - nn * mean * mean) / (nn - 1.0);
        const float sigma = (float)sqrt(var);
        stats[0] = 1.0f / (2.0f * sigma * sigma);
    }
}

// ---------------------------------------------------------------------------
// 4) sim = (E @ E^T) / max(norms_i * norms_j, 1e-8) via V_WMMA_F32_16X16X4_F32
//    One wave per 16x16 tile; 24x24 tiles; K-loop of 128 WMMAs.
//    A frag (16x4):  lane-half -> K block {0,1}/{2,3}; contiguous float2 load.
//    B frag (4x16):  B[k][n] = E[n][k]  -> same load pattern on row tileN+n.
//    C/D: VGPR r, lane L -> M = r + 8*(L>>4), N = L&15.
// ---------------------------------------------------------------------------
__global__ void pdl_gram_wmma_kernel(const float* __restrict__ E,
                                     const float* __restrict__ norms,
                                     float* __restrict__ sim) {
    const int wave = (int)((blockIdx.x * blockDim.x + threadIdx.x) >> 5);
    const int lane = (int)(threadIdx.x & 31);
    const int tm = (wave / 24) * 16;
    const int tn = (wave % 24) * 16;
    const int half = lane >> 4;       // 0 or 1: selects K block
    const int l16  = lane & 15;       // M (A) / N (B) within tile

    const float* arow = E + (size_t)(tm + l16) * PDL_D + 2 * half;
    const float* brow = E + (size_t)(tn + l16) * PDL_D + 2 * half;

    v8f c = {};
#pragma unroll 4
    for (int k = 0; k < PDL_D; k += 4) {
        v2f a = *(const v2f*)(arow + k);
        v2f b = *(const v2f*)(brow + k);
        c = __builtin_amdgcn_wmma_f32_16x16x4_f32(
                /*neg_a=*/false, a, /*neg_b=*/false, b,
                /*c_mod=*/(short)0, c, /*reuse_a=*/false, /*reuse_b=*/false);
    }

    const int gj = tn + l16;
    const float nj = norms[gj];
#pragma unroll
    for (int r = 0; r < 8; ++r) {
        const int gi = tm + r + 8 * half;
        const float denom = fmaxf(norms[gi] * nj, 1e-8f);
        sim[gi * PDL_N + gj] = c[r] / denom;
    }
}

// ---------------------------------------------------------------------------
// 5) Triplet loss, one block per anchor i.
//    With z = sim - sim_gt and flag = y_j - y_k:
//      tl = (flag < 0) ? relu(z_j - z_k) : relu(z_k - z_j)
//    Sum over k < j, j != i, k != i.
// ---------------------------------------------------------------------------
__global__ void pdl_triplet_kernel(const float* __restrict__ sim,
                                   const float* __restrict__ dist,
                                   const float* __restrict__ stats,
                                   float* __restrict__ blocksums) {
    __shared__ float zsh[PDL_N];
    __shared__ float ysh[PDL_N];
    __shared__ float wsum[PDL_N / 32];

    const int i = blockIdx.x;
    const int t = threadIdx.x;               // 384 threads = 12 waves

    const float inv2s2 = stats[0];
    const float dd = dist[i * PDL_N + t];
    const float y  = expf(-dd * dd * inv2s2);
    ysh[t] = y;
    zsh[t] = sim[i * PDL_N + t] - y;
    __syncthreads();

    float acc = 0.f;
    const int j = t;
    if (j != i) {
        const float yj = ysh[j];
        const float zj = zsh[j];
        for (int k = 0; k < j; ++k) {
            if (k == i) continue;
            const float flag = yj - ysh[k];
            const float u    = zj - zsh[k];
            acc += (flag < 0.f) ? fmaxf(u, 0.f) : fmaxf(-u, 0.f);
        }
    }

    // wave32 reduction, then cross-wave via LDS
    for (int off = 16; off > 0; off >>= 1)
        acc += __shfl_down(acc, off, 32);
    if ((t & 31) == 0) wsum[t >> 5] = acc;
    __syncthreads();
    if (t == 0) {
        float s = 0.f;
        for (int w = 0; w < PDL_N / 32; ++w) s += wsum[w];
        blocksums[i] = s;
    }
}

// ---------------------------------------------------------------------------
// 6) Final scalar: loss = sum(blocksums) / (n*(n-1)*(n-2)/2)
// ---------------------------------------------------------------------------
__global__ void pdl_final_kernel(const float* __restrict__ blocksums,
                                 float* __restrict__ out) {
    if (threadIdx.x == 0 && blockIdx.x == 0) {
        double s = 0.0;
        for (int i = 0; i < PDL_N; ++i) s += (double)blocksums[i];
        const double num_triplets = (double)PDL_N * (PDL_N - 1) * (PDL_N - 2) * 0.5;
        out[0] = (float)(s / num_triplets);
    }
}

// ---------------------------------------------------------------------------
extern "C" void kernel_launch(void* const* d_in, const int* in_sizes, int n_in,
                              void* d_out, int out_size, void* d_ws, size_t ws_size,
                              hipStream_t stream) {
    (void)in_sizes; (void)n_in; (void)out_size; (void)ws_size;

    const float* E      = (const float*)d_in[0];   // embeddings [384,512] f32
    const float* coords = (const float*)d_in[1];   // coords     [384,2]   f32
    float* out = (float*)d_out;                    // scalar f32

    // Workspace layout (floats)
    float* ws        = (float*)d_ws;
    float* dist      = ws;                                   // NN
    float* sim       = ws + PDL_NN;                          // NN
    float* norms     = ws + 2 * PDL_NN;                      // 384
    float* partials  = norms + PDL_N;                        // 2*576
    float* stats     = partials + 2 * PDL_DBLK;              // 1
    float* blocksums = stats + 4;                            // 384
    // total ~1.19 MB

    // 1) row norms
    pdl_norms_kernel<<<PDL_N, 256, 0, stream>>>(E, norms);

    // 2) haversine distances + partial sums
    pdl_dist_kernel<<<PDL_DBLK, 256, 0, stream>>>(coords, dist, partials);

    // 3) sigma -> 1/(2 sigma^2)
    pdl_sigma_kernel<<<1, 256, 0, stream>>>(partials, stats);

    // 4) sim via f32 WMMA: 24x24 tiles, 1 wave each, 4 waves/block -> 144 blocks
    pdl_gram_wmma_kernel<<<144, 128, 0, stream>>>(E, norms, sim);

    // 5) per-anchor triplet loss
    pdl_triplet_kernel<<<PDL_N, PDL_N, 0, stream>>>(sim, dist, stats, blocksums);

    // 6) final scalar
    pdl_final_kernel<<<1, 32, 0, stream>>>(blocksums, out);
}